// reaction_diffusion_nature_layer_90872918049138
// MI455X (gfx1250) — compile-verified
//
#include <hip/hip_runtime.h>
#include <math.h>

// Problem constants (match reference).
namespace {
constexpr int kB = 64;      // batches
constexpr int kC = 2;       // channels (only ch 0 used)
constexpr int kN = 1024;    // nodes
constexpr int kL = 64;      // features
constexpr int kE = 32768;   // edges
constexpr int kRES = 12;
constexpr int kT = 24;      // 288 / kRES
constexpr int kThreads = 256;        // 8 wave32 per block
constexpr int kNodesPerBlock = 128;  // grid.x = kN / kNodesPerBlock = 8
}

__device__ __forceinline__ float fast_tanh(float x) {
#if __has_builtin(__builtin_amdgcn_tanhf)
  return __builtin_amdgcn_tanhf(x);   // v_tanh_f32 on gfx1250
#else
  return tanhf(x);
#endif
}

__device__ __forceinline__ float readlane_f(float v, int k) {
  return __int_as_float(__builtin_amdgcn_readlane(__float_as_int(v), k));
}

// ---------------- CSR construction ----------------

__global__ void zero_kernel(int* __restrict__ p, int n) {
  int i = blockIdx.x * blockDim.x + threadIdx.x;
  if (i < n) p[i] = 0;
}

__global__ void count_kernel(const int* __restrict__ ei, const int* __restrict__ ej,
                             int* __restrict__ cnt_r, int* __restrict__ cnt_d) {
  int e = blockIdx.x * blockDim.x + threadIdx.x;
  if (e < kE) {
    atomicAdd(&cnt_r[ei[e]], 1);   // reaction rows grouped by edge_i
    atomicAdd(&cnt_d[ej[e]], 1);   // diffusion rows grouped by edge_j
  }
}

// One block per array (blockIdx.x: 0 = react, 1 = diff). 1024 threads.
__global__ __launch_bounds__(kN) void scan_kernel(int* __restrict__ cur_r, int* __restrict__ cur_d,
                                                  int* __restrict__ off_r, int* __restrict__ off_d) {
  __shared__ int sh[kN];
  const int t = threadIdx.x;
  int* cur = blockIdx.x ? cur_d : cur_r;
  int* off = blockIdx.x ? off_d : off_r;
  sh[t] = cur[t];
  __syncthreads();
  for (int o = 1; o < kN; o <<= 1) {          // Hillis-Steele inclusive scan
    int add = (t >= o) ? sh[t - o] : 0;
    __syncthreads();
    sh[t] += add;
    __syncthreads();
  }
  off[t + 1] = sh[t];                         // inclusive -> off[v+1]
  if (t == 0) off[0] = 0;
  cur[t] = (t == 0) ? 0 : sh[t - 1];          // exclusive prefix = fill cursor
}

__global__ void fill_kernel(const int* __restrict__ ei, const int* __restrict__ ej,
                            int* __restrict__ cur_r, int* __restrict__ cur_d,
                            int2* __restrict__ csr_r, int2* __restrict__ csr_d) {
  int e = blockIdx.x * blockDim.x + threadIdx.x;
  if (e < kE) {
    const int i = ei[e], j = ej[e];
    int sr = atomicAdd(&cur_r[i], 1);
    csr_r[sr] = make_int2(j, e);              // (neighbor col, edge id)
    int sd = atomicAdd(&cur_d[j], 1);
    csr_d[sd] = make_int2(i, e);
  }
}

// Sort each row segment by edge id -> deterministic FP summation order
// regardless of atomic fill order. Avg degree 32 -> tiny serial sorts.
__global__ void sort_rows_kernel(const int* __restrict__ off, int2* __restrict__ csr) {
  int r = blockIdx.x * blockDim.x + threadIdx.x;
  if (r >= kN) return;
  const int s0 = off[r], s1 = off[r + 1];
  for (int a = s0 + 1; a < s1; ++a) {
    int2 key = csr[a];
    int p = a - 1;
    while (p >= s0 && csr[p].y > key.y) { csr[p + 1] = csr[p]; --p; }
    csr[p + 1] = key;
  }
}

// Prebake CSR-ordered (wr, wd) pairs for each of the 24 weight tables so the
// 64-batch hot loop reads weights fully coalesced instead of random-gathering.
__global__ void reorder_weights_kernel(const int2* __restrict__ csr_r, const int2* __restrict__ csr_d,
                                       const float* __restrict__ wreact, const float* __restrict__ wdiff,
                                       float2* __restrict__ wrd_r, float2* __restrict__ wrd_d) {
  const int slot = blockIdx.x * blockDim.x + threadIdx.x;
  const int t = blockIdx.y;
  if (slot < kE) {
    const float* wr = wreact + (size_t)t * kE;
    const float* wd = wdiff  + (size_t)t * kE;
    int er = csr_r[slot].y;
    wrd_r[(size_t)t * kE + slot] = make_float2(wr[er], wd[er]);
    int ed = csr_d[slot].y;
    wrd_d[(size_t)t * kE + slot] = make_float2(wr[ed], wd[ed]);
  }
}

// ---------------- main fused kernel ----------------

extern __shared__ float xlds[];   // kN*kL floats = 256 KB (<= 320 KB WGP LDS)

template <bool PRE>
__global__ __launch_bounds__(kThreads, 1)
void rd_main(const float* __restrict__ inputs, const int* __restrict__ ind,
             const float* __restrict__ wreact, const float* __restrict__ wdiff,
             const float* __restrict__ brea, const float* __restrict__ bdif,
             const int* __restrict__ off_r, const int* __restrict__ off_d,
             const int2* __restrict__ csr_r, const int2* __restrict__ csr_d,
             const float2* __restrict__ wrd_r, const float2* __restrict__ wrd_d,
             float* __restrict__ out) {
  const int b  = blockIdx.y;
  const int ii = __builtin_amdgcn_readfirstlane(ind[b]) / kRES;  // table index, [0,24)
  const float* xb = inputs + (size_t)b * (kC * kN * kL);         // channel 0
  const int tid = threadIdx.x;

  // Stage the full x[b] (1024x64 f32 = 256 KB) into LDS via CDNA5
  // async global->LDS DMA (tracked by ASYNCcnt, waited explicitly).
  for (int i = tid; i < (kN * kL) / 4; i += kThreads) {
    unsigned lds_off = (unsigned)(size_t)(xlds + 4 * i);          // LDS byte addr
    unsigned long long gaddr = (unsigned long long)(const void*)(xb + 4 * i);
    asm volatile("global_load_async_to_lds_b128 %0, %1, off"
                 :: "v"(lds_off), "v"(gaddr) : "memory");
  }
  asm volatile("s_wait_asynccnt 0" ::: "memory");
  __syncthreads();

  const float*  wr_row = wreact + (size_t)ii * kE;
  const float*  wd_row = wdiff  + (size_t)ii * kE;
  const float2* pr_row = wrd_r + (size_t)ii * kE;   // prebaked (wr,wd) in csr_r order
  const float2* pd_row = wrd_d + (size_t)ii * kE;   // prebaked (wr,wd) in csr_d order
  const int wave = tid >> 5;
  const int lane = tid & 31;

  for (int n = wave; n < kNodesPerBlock; n += (kThreads / 32)) {
    const int w = blockIdx.x * kNodesPerBlock + n;
    float aR0 = 0.f, aR1 = 0.f, aD0 = 0.f, aD1 = 0.f;
    float degD = 0.f, degR = 0.f;

    // Reaction row: edges with edge_i == w. Gather x[edge_j] * wr.
    // Same edge set also supplies deg_d(w) = sum wd.
    int s0 = __builtin_amdgcn_readfirstlane(off_r[w]);
    int s1 = __builtin_amdgcn_readfirstlane(off_r[w + 1]);
    for (int s = s0; s < s1; s += 32) {
      int m = s1 - s; if (m > 32) m = 32;
      __builtin_prefetch(&csr_r[s + 32], 0, 0);     // global_prefetch_b8
      int col = 0; float vr = 0.f, vd = 0.f;
      if (lane < m) {
        int2 ce = csr_r[s + lane];                  // coalesced per-lane load
        col = ce.x;
        if (PRE) { float2 wv2 = pr_row[s + lane]; vr = wv2.x; vd = wv2.y; }
        else     { vr = wr_row[ce.y]; vd = wd_row[ce.y]; }
      }
      degD += vd;
      for (int k = 0; k < m; ++k) {                 // wave-uniform broadcast via v_readlane
        int   c  = __builtin_amdgcn_readlane(col, k);
        float wv = readlane_f(vr, k);
        const float2 xr = *(const float2*)(xlds + c * kL + 2 * lane);  // ds_load_b64, 64 banks
        aR0 = fmaf(wv, xr.x, aR0);
        aR1 = fmaf(wv, xr.y, aR1);
      }
    }

    // Diffusion row: edges with edge_j == w. Gather x[edge_i] * wd.
    // Same edge set supplies deg_r(w) = sum wr.
    s0 = __builtin_amdgcn_readfirstlane(off_d[w]);
    s1 = __builtin_amdgcn_readfirstlane(off_d[w + 1]);
    for (int s = s0; s < s1; s += 32) {
      int m = s1 - s; if (m > 32) m = 32;
      __builtin_prefetch(&csr_d[s + 32], 0, 0);
      int col = 0; float vr = 0.f, vd = 0.f;
      if (lane < m) {
        int2 ce = csr_d[s + lane];
        col = ce.x;
        if (PRE) { float2 wv2 = pd_row[s + lane]; vr = wv2.x; vd = wv2.y; }
        else     { vr = wr_row[ce.y]; vd = wd_row[ce.y]; }
      }
      degR += vr;
      for (int k = 0; k < m; ++k) {
        int   c  = __builtin_amdgcn_readlane(col, k);
        float wv = readlane_f(vd, k);
        const float2 xr = *(const float2*)(xlds + c * kL + 2 * lane);
        aD0 = fmaf(wv, xr.x, aD0);
        aD1 = fmaf(wv, xr.y, aD1);
      }
    }

    // Wave-wide reduce of the degree partial sums.
    for (int o = 16; o > 0; o >>= 1) {
      degD += __shfl_xor(degD, o);
      degR += __shfl_xor(degR, o);
    }

    const float2 xw = *(const float2*)(xlds + w * kL + 2 * lane);
    const float br  = brea[ii * kN + w];
    const float bd  = bdif[ii * kN + w];

    const float r0 = fmaf(degR, xw.x, br) - aR0;   // reaction + bias
    const float r1 = fmaf(degR, xw.y, br) - aR1;
    const float d0 = fmaf(degD, xw.x, bd) - aD0;   // diffusion + bias
    const float d1 = fmaf(degD, xw.y, bd) - aD1;

    float2 o2;
    o2.x = fast_tanh(r0) + d0 + xw.x;
    o2.y = fast_tanh(r1) + d1 + xw.y;
    *(float2*)(out + ((size_t)b * kN + w) * kL + 2 * lane) = o2;  // 256B/wave store
  }
}

// ---------------- launcher ----------------

extern "C" void kernel_launch(void* const* d_in, const int* in_sizes, int n_in,
                              void* d_out, int out_size, void* d_ws, size_t ws_size,
                              hipStream_t stream) {
  (void)in_sizes; (void)n_in; (void)out_size;
  const float* inputs  = (const float*)d_in[0];
  const int*   ind     = (const int*)d_in[1];
  const int*   edge_i  = (const int*)d_in[2];
  const int*   edge_j  = (const int*)d_in[3];
  const float* w_react = (const float*)d_in[4];
  const float* w_diff  = (const float*)d_in[5];
  const float* b_react = (const float*)d_in[6];
  const float* b_diff  = (const float*)d_in[7];
  float* out = (float*)d_out;

  // Workspace layout. Base: ~0.55 MB. Prebaked weights: +12.6 MB (optional).
  int*  off_r = (int*)d_ws;            // kN+1
  int*  off_d = off_r + (kN + 1);      // kN+1
  int*  cur_r = off_d + (kN + 1);      // kN (counts, then cursors)
  int*  cur_d = cur_r + kN;            // kN
  int2* csr_r = (int2*)(cur_d + kN);   // kE   (8B-aligned: 16392 % 8 == 0)
  int2* csr_d = csr_r + kE;            // kE
  float2* wrd_r = (float2*)(csr_d + kE);        // kT*kE
  float2* wrd_d = wrd_r + (size_t)kT * kE;      // kT*kE
  const size_t ws_needed_pre =
      (size_t)(2 * (kN + 1) + 2 * kN) * sizeof(int) +
      (size_t)2 * kE * sizeof(int2) + (size_t)2 * kT * kE * sizeof(float2);
  const bool pre = ws_size >= ws_needed_pre;    // deterministic per-run choice

  zero_kernel<<<dim3((2 * kN + kThreads - 1) / kThreads), kThreads, 0, stream>>>(cur_r, 2 * kN);
  count_kernel<<<dim3(kE / kThreads), kThreads, 0, stream>>>(edge_i, edge_j, cur_r, cur_d);
  scan_kernel<<<dim3(2), kN, 0, stream>>>(cur_r, cur_d, off_r, off_d);
  fill_kernel<<<dim3(kE / kThreads), kThreads, 0, stream>>>(edge_i, edge_j, cur_r, cur_d, csr_r, csr_d);
  sort_rows_kernel<<<dim3(kN / kThreads), kThreads, 0, stream>>>(off_r, csr_r);
  sort_rows_kernel<<<dim3(kN / kThreads), kThreads, 0, stream>>>(off_d, csr_d);
  if (pre) {
    reorder_weights_kernel<<<dim3(kE / kThreads, kT), kThreads, 0, stream>>>(
        csr_r, csr_d, w_react, w_diff, wrd_r, wrd_d);
  }

  const size_t shmem = (size_t)kN * kL * sizeof(float);  // 256 KB dynamic LDS
  if (pre) {
    (void)hipFuncSetAttribute(reinterpret_cast<const void*>(rd_main<true>),
                              hipFuncAttributeMaxDynamicSharedMemorySize, (int)shmem);
    rd_main<true><<<dim3(kN / kNodesPerBlock, kB), kThreads, shmem, stream>>>(
        inputs, ind, w_react, w_diff, b_react, b_diff,
        off_r, off_d, csr_r, csr_d, wrd_r, wrd_d, out);
  } else {
    (void)hipFuncSetAttribute(reinterpret_cast<const void*>(rd_main<false>),
                              hipFuncAttributeMaxDynamicSharedMemorySize, (int)shmem);
    rd_main<false><<<dim3(kN / kNodesPerBlock, kB), kThreads, shmem, stream>>>(
        inputs, ind, w_react, w_diff, b_react, b_diff,
        off_r, off_d, csr_r, csr_d, wrd_r, wrd_d, out);
  }
}